// HAModule_75814762709628
// MI455X (gfx1250) — compile-verified
//
#include <hip/hip_runtime.h>

// Problem constants (match reference).
#define N_UP   100000
#define N_DOWN 500000
#define DIM    128

typedef __attribute__((ext_vector_type(2))) float v2f;
typedef __attribute__((ext_vector_type(4))) float f32x4;
typedef __attribute__((ext_vector_type(8))) float v8f;

// ---------------------------------------------------------------------------
// Kernel A: zero the segment-sum accumulator (aliased onto out_up region of
// d_out) and the per-parent counts (in d_ws). 51.2MB + 400KB, L2-resident.
// ---------------------------------------------------------------------------
__global__ void ha_zero_kernel(f32x4* __restrict__ seg4, float* __restrict__ cnt) {
    const int i = blockIdx.x * blockDim.x + threadIdx.x;
    const int nseg4 = N_UP * DIM / 4;  // 3.2M float4
    if (i < nseg4) seg4[i] = (f32x4){0.f, 0.f, 0.f, 0.f};
    if (i < N_UP)  cnt[i] = 0.f;
}

// ---------------------------------------------------------------------------
// Kernel B: streaming pass over emb_down. One wave32 per row (32 lanes x
// float4 = 512B = one row).
//  - NT load of emb_down (read-once stream, keep L2 for emb_up/seg)
//  - gather emb_up[parent] (L2-resident, scalar base), residual add,
//    NT store out_down
//  - fire-and-forget global_atomic_add_f32 into seg accumulator + count
// ---------------------------------------------------------------------------
__global__ void ha_scatter_kernel(const float* __restrict__ emb_up,
                                  const float* __restrict__ emb_down,
                                  const int*   __restrict__ parent,
                                  float* __restrict__ out_down,
                                  float* __restrict__ seg,
                                  float* __restrict__ cnt) {
    const int wave = threadIdx.x >> 5;       // 8 waves / block
    const int lane = threadIdx.x & 31;
    const int row  = blockIdx.x * 8 + wave;  // N_DOWN % 8 == 0, no tail

    // Prefetch the streaming input a few rows ahead (global_prefetch_b8).
    if (row + 64 < N_DOWN)
        __builtin_prefetch(emb_down + (size_t)(row + 64) * DIM + lane * 4, 0, 1);

    // Parent index is wave-uniform: force it scalar for SADDR-based gathers.
    int p = parent[row];
    p = __builtin_amdgcn_readfirstlane(p);

    const f32x4* dn = (const f32x4*)(emb_down + (size_t)row * DIM);
    f32x4 v = __builtin_nontemporal_load(&dn[lane]);           // stream in (NT)

    const f32x4* up = (const f32x4*)(emb_up + (size_t)p * DIM);
    f32x4 u = up[lane];                                        // L2-resident gather

    f32x4 o = {v.x + u.x, v.y + u.y, v.z + u.z, v.w + u.w};
    __builtin_nontemporal_store(o, (f32x4*)(out_down + (size_t)row * DIM) + lane);

    float* s = seg + (size_t)p * DIM + lane * 4;               // L2-resident atomics
    atomicAdd(s + 0, v.x);
    atomicAdd(s + 1, v.y);
    atomicAdd(s + 2, v.z);
    atomicAdd(s + 3, v.w);
    if (lane == 0) atomicAdd(&cnt[p], 1.0f);
}

// ---------------------------------------------------------------------------
// Kernel C: out_up = emb_up + diag(1/cnt) x seg, per 16x16 tile on the matrix
// unit via 4 chained V_WMMA_F32_16X16X4_F32. The mean normalization is folded
// into the diagonal A operand: since A[m][k] is nonzero only at k==m, each
// lane contributes exactly one value inv[row0 + (lane&15)] -> ONE division
// per lane total (vs 16 per-element divides), and B is raw seg loads.
//
// Operand layouts (CDNA5 ISA):
//   A 16x4 : lanes 0-15  -> M=lane,    VGPR0=K0, VGPR1=K1
//            lanes 16-31 -> M=lane-16, VGPR0=K2, VGPR1=K3
//   B 4x16 : VGPR0: lanes0-15 K=k0,   lanes16-31 K=k0+2 ; N = lane&15
//            VGPR1: lanes0-15 K=k0+1, lanes16-31 K=k0+3
//   C/D    : VGPR r: lanes0-15 M=r, lanes16-31 M=r+8 ; N = lane&15
// EXEC is all-1s: exact grid, no divergence-creating control flow.
// ---------------------------------------------------------------------------
__global__ void ha_finalize_kernel(const float* __restrict__ emb_up,
                                   float* __restrict__ out_up,   // holds seg on entry
                                   const float* __restrict__ cnt) {
    const int wave = threadIdx.x >> 5;   // 0..7 -> column tile
    const int lane = threadIdx.x & 31;
    const int row0 = blockIdx.x * 16;    // N_UP/16 = 6250 blocks, exact
    const int col0 = wave * 16;          // 8 waves * 16 = 128 = DIM
    const int n    = lane & 15;
    const int hi   = lane >> 4;          // lane half
    const int m    = n;                  // this lane's A-matrix row

    // The single normalization factor this lane ever feeds into A.
    const float inv = 1.0f / fmaxf(cnt[row0 + m], 1.0f);

    // C = emb_up tile
    v8f c;
#pragma unroll
    for (int r = 0; r < 8; ++r) {
        const int mr = r + hi * 8;
        c[r] = emb_up[(size_t)(row0 + mr) * DIM + col0 + n];
    }

    // Accumulate diag(inv) x seg in K-chunks of 4.
#pragma unroll
    for (int k0 = 0; k0 < 16; k0 += 4) {
        const int kbase = k0 + hi * 2;   // this lane-half's K pair

        v2f a;                            // diagonal chunk: nonzero only at k==m
        a[0] = (m == kbase)     ? inv : 0.0f;
        a[1] = (m == kbase + 1) ? inv : 0.0f;

        v2f b;                            // raw seg rows (pooled = A x B)
        b[0] = out_up[(size_t)(row0 + kbase)     * DIM + col0 + n];
        b[1] = out_up[(size_t)(row0 + kbase + 1) * DIM + col0 + n];

        // D = A x B + C  (8 args: neg_a, A, neg_b, B, c_mod, C, reuse_a, reuse_b)
        c = __builtin_amdgcn_wmma_f32_16x16x4_f32(
                false, a, false, b, (short)0, c, false, false);
    }

    // Store finished out_up tile (overwrites the seg accumulator in place).
#pragma unroll
    for (int r = 0; r < 8; ++r) {
        const int mr = r + hi * 8;
        out_up[(size_t)(row0 + mr) * DIM + col0 + n] = c[r];
    }
}

// ---------------------------------------------------------------------------
// Launch: inputs in setup_inputs() order: emb_up(f32), emb_down(f32),
// parent_idx(int). Output: [out_up | out_down] concatenated fp32.
// d_ws holds only the 400KB count array; seg aliases the out_up region.
// ---------------------------------------------------------------------------
extern "C" void kernel_launch(void* const* d_in, const int* in_sizes, int n_in,
                              void* d_out, int out_size, void* d_ws, size_t ws_size,
                              hipStream_t stream) {
    const float* emb_up   = (const float*)d_in[0];
    const float* emb_down = (const float*)d_in[1];
    const int*   parent   = (const int*)d_in[2];

    float* out_up   = (float*)d_out;
    float* out_down = out_up + (size_t)N_UP * DIM;
    float* cnt      = (float*)d_ws;            // N_UP floats

    // A: zero seg (aliased on out_up) + counts
    const int nseg4 = N_UP * DIM / 4;
    ha_zero_kernel<<<(nseg4 + 255) / 256, 256, 0, stream>>>((f32x4*)out_up, cnt);

    // B: streaming scatter/gather pass, 8 rows per 256-thread block
    ha_scatter_kernel<<<N_DOWN / 8, 256, 0, stream>>>(
        emb_up, emb_down, parent, out_down, out_up, cnt);

    // C: WMMA finalize, one 16-row tile strip per block (8 col tiles = 8 waves)
    ha_finalize_kernel<<<N_UP / 16, 256, 0, stream>>>(emb_up, out_up, cnt);
}